// BilinearSampler_14817637171454
// MI455X (gfx1250) — compile-verified
//
#include <hip/hip_runtime.h>
#include <stdint.h>

// Problem constants (match setup_inputs exactly)
#define BB     4
#define NPTS   4096
#define DCH    256
#define HM     512
#define WM     512
#define PLANE  (HM * WM)   // 262144 elements per channel plane

// ((p/512)*2 - 1)  ->  ((g+1)*512 - 1)*0.5, replicated op-for-op in fp32
__device__ __forceinline__ float unnormalize(float p) {
    float g = __fsub_rn(__fmul_rn(__fmul_rn(p, 1.0f / 512.0f), 2.0f), 1.0f);
    return __fmul_rn(__fsub_rn(__fmul_rn(__fadd_rn(g, 1.0f), 512.0f), 1.0f), 0.5f);
}

// zero-padded corner fetch: feat plane fb[H=512][W=512], (yy -> H axis, xx -> W axis)
__device__ __forceinline__ float fetch(const float* __restrict__ fb, int yy, int xx) {
    if (xx >= 0 && xx < WM && yy >= 0 && yy < HM)
        return fb[yy * WM + xx];
    return 0.0f;
}

// F.grid_sample bilinear, align_corners=False, zero padding; weight-product-first
__device__ __forceinline__ float bilin(const float* __restrict__ fb, float ix, float iy) {
    float x0f = floorf(ix), y0f = floorf(iy);
    float wx1 = __fsub_rn(ix, x0f);
    float wy1 = __fsub_rn(iy, y0f);
    float wx0 = __fsub_rn(1.0f, wx1);
    float wy0 = __fsub_rn(1.0f, wy1);
    int x0 = (int)x0f, y0 = (int)y0f;
    float v00 = fetch(fb, y0,     x0);
    float v01 = fetch(fb, y0,     x0 + 1);
    float v10 = fetch(fb, y0 + 1, x0);
    float v11 = fetch(fb, y0 + 1, x0 + 1);
    float r = __fmul_rn(v00, __fmul_rn(wx0, wy0));
    r = __fadd_rn(r, __fmul_rn(v01, __fmul_rn(wx1, wy0)));
    r = __fadd_rn(r, __fmul_rn(v10, __fmul_rn(wx0, wy1)));
    r = __fadd_rn(r, __fmul_rn(v11, __fmul_rn(wx1, wy1)));
    return r;
}

// One block per (batch, point): lanes 0..15 async-gather the 16 mask candidates
// into LDS via the gfx1250 async-to-LDS engine, lane 0 does first-max argmax,
// then all 256 threads (one per channel) do both bilinear samples and stream
// the results out with non-temporal stores.
__global__ __launch_bounds__(256) void BilinearSampler_kernel(
    const float* __restrict__ feat,     // [B,256,512,512]
    const float* __restrict__ sp,       // [B,4096,2] (x,y)
    const float* __restrict__ mask,     // [B,2,512,512]
    float* __restrict__ out_tgt,        // [B,4096,256]
    float* __restrict__ out_pt,         // [B,4096,2]
    float* __restrict__ out_src)        // [B,4096,256]
{
    __shared__ float s_m0[16];
    __shared__ float s_m1[16];
    __shared__ float s_b[4];            // ixt, iyt, ixs, iys

    const int blk = blockIdx.x;
    const int b   = blk >> 12;          // / 4096
    const int n   = blk & (NPTS - 1);
    const int tid = threadIdx.x;

    if (tid < 16) {
        const float x = sp[(size_t)(b * NPTS + n) * 2 + 0];
        const float y = sp[(size_t)(b * NPTS + n) * 2 + 1];
        // clip(x, 0, 512) then int-truncate (reference semantics)
        const int xi = (int)fminf(fmaxf(x, 0.0f), 512.0f);
        const int yi = (int)fminf(fmaxf(y, 0.0f), 512.0f);

        // candidate k = tid: offsets arange(-2,2) x arange(-2,2), row-major
        const int oi = (tid >> 2) - 2;
        const int oj = (tid & 3) - 2;
        const int cx = xi + oi, cy = yi + oj;
        const int cxc = min(max(cx, 0), HM - 1);
        const int cyc = min(max(cy, 0), WM - 1);

        const float* p0 = mask + (size_t)b * 2 * PLANE + (size_t)cxc * WM + cyc;
        uint64_t a0 = (uint64_t)(uintptr_t)p0;
        uint64_t a1 = a0 + (uint64_t)PLANE * sizeof(float);
        uint32_t l0 = (uint32_t)(uintptr_t)&s_m0[tid];
        uint32_t l1 = (uint32_t)(uintptr_t)&s_m1[tid];
        // gfx1250 async global->LDS gather (ASYNCcnt), then wait for landing
        asm volatile(
            "global_load_async_to_lds_b32 %0, %1, off\n\t"
            "global_load_async_to_lds_b32 %2, %3, off\n\t"
            "s_wait_asynccnt 0"
            :: "v"(l0), "v"(a0), "v"(l1), "v"(a1)
            : "memory");

        if (tid == 0) {
            float best_s = -__builtin_inff();
            int   best_k = 0;
            #pragma unroll
            for (int k = 0; k < 16; ++k) {
                const int koi = (k >> 2) - 2;
                const int koj = (k & 3) - 2;
                const int kcx = xi + koi, kcy = yi + koj;
                const bool valid = (kcx >= 0) && (kcx < HM) && (kcy >= 0) && (kcy < WM);
                const float within = ((koi * koi + koj * koj) <= 4) ? 1.0f : 0.0f;
                float sc = __fmul_rn(__fadd_rn(s_m0[k], s_m1[k]), within);
                sc = valid ? sc : -__builtin_inff();
                if (sc > best_s) { best_s = sc; best_k = k; }   // first max wins
            }
            float px = (float)(xi + ((best_k >> 2) - 2));
            float py = (float)(yi + ((best_k & 3) - 2));
            if (x == 0.0f && y == 0.0f) { px = x; py = y; }     // (0,0) pass-through
            out_pt[(size_t)(b * NPTS + n) * 2 + 0] = px;
            out_pt[(size_t)(b * NPTS + n) * 2 + 1] = py;
            s_b[0] = unnormalize(px);
            s_b[1] = unnormalize(py);
            s_b[2] = unnormalize(x);
            s_b[3] = unnormalize(y);
        }
    }
    __syncthreads();

    const float ixt = s_b[0], iyt = s_b[1];
    const float ixs = s_b[2], iys = s_b[3];

    // one channel per thread; channel planes are 1 MiB apart (gather-bound)
    const float* fb = feat + (size_t)(b * DCH + tid) * PLANE;
    const float tv = bilin(fb, ixt, iyt);
    const float sv = bilin(fb, ixs, iys);

    const size_t o = (size_t)(b * NPTS + n) * DCH + tid;
    // outputs are write-once streams: non-temporal so they don't evict
    // reusable feature-map lines from the 192 MB L2
    __builtin_nontemporal_store(tv, out_tgt + o);
    __builtin_nontemporal_store(sv, out_src + o);
}

extern "C" void kernel_launch(void* const* d_in, const int* in_sizes, int n_in,
                              void* d_out, int out_size, void* d_ws, size_t ws_size,
                              hipStream_t stream) {
    (void)in_sizes; (void)n_in; (void)d_ws; (void)ws_size; (void)out_size;
    const float* feat = (const float*)d_in[0];   // [4,256,512,512]
    const float* sp   = (const float*)d_in[1];   // [4,4096,2]
    const float* mask = (const float*)d_in[2];   // [4,2,512,512]

    float* out     = (float*)d_out;
    float* out_tgt = out;                                        // [4,4096,256]
    float* out_pt  = out_tgt + (size_t)BB * NPTS * DCH;          // [4,4096,2]
    float* out_src = out_pt  + (size_t)BB * NPTS * 2;            // [4,4096,256]

    dim3 grid(BB * NPTS);
    dim3 block(DCH);
    hipLaunchKernelGGL(BilinearSampler_kernel, grid, block, 0, stream,
                       feat, sp, mask, out_tgt, out_pt, out_src);
}